// sparse_Mat_Layer_15530601742411
// MI455X (gfx1250) — compile-verified
//
#include <hip/hip_runtime.h>
#include <stdint.h>

#define BLOCK 256
#define EPT   8                      // nonzeros per thread per tile
#define TILE  (BLOCK * EPT)          // 2048 nonzeros per tile
#define MAXGRID 1024                 // persistent blocks

// ---------------------------------------------------------------------------
// Helpers
// ---------------------------------------------------------------------------

// Low 32 bits of a generic pointer into LDS == LDS byte offset.
__device__ __forceinline__ uint32_t lds_addr32(const void* p) {
    return (uint32_t)(uintptr_t)p;
}

// Async DMA 16B global->LDS, non-temporal (streaming data must not evict the
// L2-hot x/y vectors). Tracked by ASYNCcnt.
__device__ __forceinline__ void async_copy_b128_nt(uint32_t lds_byte_addr, const void* gaddr) {
    asm volatile("global_load_async_to_lds_b128 %0, %1, off th:TH_LOAD_NT"
                 :: "v"(lds_byte_addr), "v"(gaddr)
                 : "memory");
}

__device__ __forceinline__ void wait_asynccnt_le6() {
    asm volatile("s_wait_asynccnt 0x6" ::: "memory");
}
__device__ __forceinline__ void wait_asynccnt_0() {
    asm volatile("s_wait_asynccnt 0x0" ::: "memory");
}

// Native f32 global atomic add (global_atomic_add_f32, no CAS loop).
__device__ __forceinline__ void atomic_add_f32(float* p, float v) {
    unsafeAtomicAdd(p, v);
}

// ---------------------------------------------------------------------------
// Kernel 1: zero the output, warm x into L2 (global_prefetch_b8)
// ---------------------------------------------------------------------------
__global__ void __launch_bounds__(BLOCK)
zero_and_warm(float* __restrict__ y, int ny, const float* __restrict__ x, int nx) {
    const int i = blockIdx.x * BLOCK + threadIdx.x;
    if (i < ny) y[i] = 0.0f;
    const int pf = i * 32;                  // one prefetch per 128B region
    if (pf < nx) __builtin_prefetch(&x[pf], 0, 1);
}

// ---------------------------------------------------------------------------
// Kernel 2: COO SpMV; vals/rows/cols streamed through LDS with double-buffered
// async copies; gathers of x hit L2; scatter via native f32 atomics.
// ---------------------------------------------------------------------------
__global__ void __launch_bounds__(BLOCK)
spmv_coo_async(const float* __restrict__ x,
               const float* __restrict__ vals,
               const int*   __restrict__ rows,
               const int*   __restrict__ cols,
               float*       __restrict__ y,
               int ntiles)
{
    __shared__ __align__(16) float s_vals[2][TILE];
    __shared__ __align__(16) int   s_rows[2][TILE];
    __shared__ __align__(16) int   s_cols[2][TILE];

    const int t = threadIdx.x;
    const int e = t * EPT;                       // this thread's 8-element slot

    // LDS byte addresses of this thread's slot in buffer 0; buffer 1 is at a
    // fixed +TILE*4 byte stride within each array.
    const uint32_t bufstride = (uint32_t)TILE * 4u;
    const uint32_t lv0 = lds_addr32(&s_vals[0][e]);
    const uint32_t lr0 = lds_addr32(&s_rows[0][e]);
    const uint32_t lc0 = lds_addr32(&s_cols[0][e]);

    int tile = blockIdx.x;
    if (tile >= ntiles) return;

    // Issue one tile's copies: 2 x b128 per array per thread (8 elements).
    auto issue = [&](int tl, int buf) {
        const size_t   base = (size_t)tl * TILE + e;
        const uint32_t bofs = (uint32_t)buf * bufstride;
        async_copy_b128_nt(lv0 + bofs,      vals + base);
        async_copy_b128_nt(lv0 + bofs + 16, vals + base + 4);
        async_copy_b128_nt(lr0 + bofs,      rows + base);
        async_copy_b128_nt(lr0 + bofs + 16, rows + base + 4);
        async_copy_b128_nt(lc0 + bofs,      cols + base);
        async_copy_b128_nt(lc0 + bofs + 16, cols + base + 4);
    };

    issue(tile, 0);                              // prologue
    int buf = 0;
    const int stride = gridDim.x;
    for (; tile < ntiles; tile += stride) {
        const int nxt = tile + stride;
        if (nxt < ntiles) {
            issue(nxt, buf ^ 1);                 // prefetch next tile
            wait_asynccnt_le6();                 // older tile's 6 copies landed
        } else {
            wait_asynccnt_0();
        }
        __syncthreads();                         // whole tile visible in LDS

        const float4 v0 = *reinterpret_cast<const float4*>(&s_vals[buf][e]);
        const float4 v1 = *reinterpret_cast<const float4*>(&s_vals[buf][e + 4]);
        const int4   r0 = *reinterpret_cast<const int4*>(&s_rows[buf][e]);
        const int4   r1 = *reinterpret_cast<const int4*>(&s_rows[buf][e + 4]);
        const int4   c0 = *reinterpret_cast<const int4*>(&s_cols[buf][e]);
        const int4   c1 = *reinterpret_cast<const int4*>(&s_cols[buf][e + 4]);

        // 8 independent gathers (L2-hot x), then 8 atomics — maximize MLP.
        const float x0 = x[c0.x], x1 = x[c0.y], x2 = x[c0.z], x3 = x[c0.w];
        const float x4 = x[c1.x], x5 = x[c1.y], x6 = x[c1.z], x7 = x[c1.w];

        atomic_add_f32(&y[r0.x], v0.x * x0);
        atomic_add_f32(&y[r0.y], v0.y * x1);
        atomic_add_f32(&y[r0.z], v0.z * x2);
        atomic_add_f32(&y[r0.w], v0.w * x3);
        atomic_add_f32(&y[r1.x], v1.x * x4);
        atomic_add_f32(&y[r1.y], v1.y * x5);
        atomic_add_f32(&y[r1.z], v1.z * x6);
        atomic_add_f32(&y[r1.w], v1.w * x7);

        __syncthreads();                         // done reading before overwrite
        buf ^= 1;
    }
}

// ---------------------------------------------------------------------------
// Kernel 3: scalar tail for nnz not divisible by TILE
// ---------------------------------------------------------------------------
__global__ void __launch_bounds__(BLOCK)
spmv_coo_tail(const float* __restrict__ x,
              const float* __restrict__ vals,
              const int*   __restrict__ rows,
              const int*   __restrict__ cols,
              float*       __restrict__ y,
              int start, int nnz)
{
    const int i = start + blockIdx.x * BLOCK + threadIdx.x;
    if (i < nnz) {
        const float v = __builtin_nontemporal_load(&vals[i]);
        const int   r = __builtin_nontemporal_load(&rows[i]);
        const int   c = __builtin_nontemporal_load(&cols[i]);
        atomic_add_f32(&y[r], v * x[c]);
    }
}

// ---------------------------------------------------------------------------
// Launch
// ---------------------------------------------------------------------------
extern "C" void kernel_launch(void* const* d_in, const int* in_sizes, int n_in,
                              void* d_out, int out_size, void* d_ws, size_t ws_size,
                              hipStream_t stream) {
    const float* x    = (const float*)d_in[0];
    const float* vals = (const float*)d_in[1];
    const int*   rows = (const int*)d_in[2];
    const int*   cols = (const int*)d_in[3];
    float*       y    = (float*)d_out;

    const int nx  = in_sizes[0];
    const int nnz = in_sizes[1];

    // Zero output (harness poisons it) + warm x into L2.
    {
        const int grid = (out_size + BLOCK - 1) / BLOCK;
        zero_and_warm<<<grid, BLOCK, 0, stream>>>(y, out_size, x, nx);
    }

    const int ntiles = nnz / TILE;
    if (ntiles > 0) {
        const int grid = ntiles < MAXGRID ? ntiles : MAXGRID;
        spmv_coo_async<<<grid, BLOCK, 0, stream>>>(x, vals, rows, cols, y, ntiles);
    }
    const int start = ntiles * TILE;
    if (start < nnz) {
        const int rem  = nnz - start;
        const int grid = (rem + BLOCK - 1) / BLOCK;
        spmv_coo_tail<<<grid, BLOCK, 0, stream>>>(x, vals, rows, cols, y, start, nnz);
    }
}